// OhemCELoss_9010841387760
// MI455X (gfx1250) — compile-verified
//
#include <hip/hip_runtime.h>
#include <hip/hip_bf16.h>
#include <stdint.h>

// ---------------------------------------------------------------------------
// OHEM cross-entropy, MI455X (gfx1250, wave32).
// Bandwidth-bound: 318MB logits streamed once (NT b128 loads), 16MB loss array
// kept in the 192MB L2 for a 4x8-bit radix-select of the rank-N_MIN value,
// then one deterministic counting/summing pass.
// CDNA5 paths: nontemporal b128 loads, global_prefetch_b8, and
// GLOBAL_LOAD_ASYNC_TO_LDS_B128 + s_wait_asynccnt double-buffering for the
// L2-resident passes. No GEMM structure -> no WMMA.
// ---------------------------------------------------------------------------

typedef float v4f __attribute__((ext_vector_type(4)));

#define NPIX    (16 * 512 * 512)      // 4194304 pixels
#define NG      (NPIX / 4)            // float4 groups = 1048576
#define HW      (512 * 512)           // 262144
#define HW4     (HW / 4)              // 65536
#define NC      19
#define NMIN    262144
#define THRESH  0.3566749394f         // float32(-log(0.7)), matches numpy cast
#define IGN     255
#define NBLK    1024                  // blocks for streaming passes (fixed -> deterministic)

// workspace layout (bytes); requires ~16.03 MB of d_ws
#define LOSS_OFF   0
#define HIST_OFF   (NPIX * 4)                 // 4 passes x 256 uint bins
#define STATE_OFF  (HIST_OFF + 4 * 256 * 4)
#define PART_OFF   (STATE_OFF + 64)

struct SelState {
    unsigned int prefix;   // selected key bits so far (MSB-first)
    unsigned int rank;     // remaining 0-indexed descending rank
    float        v;        // final: exact rank-NMIN loss value
};

static __device__ inline v4f v4max(v4f a, v4f b) {
    v4f r; r.x = fmaxf(a.x, b.x); r.y = fmaxf(a.y, b.y);
    r.z = fmaxf(a.z, b.z); r.w = fmaxf(a.w, b.w); return r;
}
static __device__ inline v4f v4exp(v4f a) {
    v4f r; r.x = __expf(a.x); r.y = __expf(a.y);
    r.z = __expf(a.z); r.w = __expf(a.w); return r;
}

// ---------------- CDNA5 async global->LDS path (guarded, with fallback) -----
#if __has_builtin(__builtin_amdgcn_global_load_async_to_lds_b128)
#define HAVE_ASYNC 1
// Param 1 (per hipcc diagnostic): 'int __vector(4) __device__ *'  == AS1 v4i*
typedef int v4i __attribute__((vector_size(16)));
typedef __attribute__((address_space(1))) v4i* gv4p;   // global (AS1) int4*
typedef __attribute__((address_space(3))) v4i* lv4p;   // LDS    (AS3) int4*
static __device__ inline void async_ld16(const void* g, void* l) {
    // (global src, lds dst, imm offset, imm cpol)
    __builtin_amdgcn_global_load_async_to_lds_b128((gv4p)g, (lv4p)l, 0, 0);
}
static __device__ inline void wait_async(int n) {
#if __has_builtin(__builtin_amdgcn_s_wait_asynccnt)
    if (n == 0) __builtin_amdgcn_s_wait_asynccnt(0);
    else        __builtin_amdgcn_s_wait_asynccnt(1);
#else
    if (n == 0) asm volatile("s_wait_asynccnt 0x0" ::: "memory");
    else        asm volatile("s_wait_asynccnt 0x1" ::: "memory");
#endif
}
#endif

// ---------------------------------------------------------------- init ------
__global__ __launch_bounds__(256) void k_init(unsigned int* __restrict__ hist,
                                              SelState* __restrict__ st) {
    int i = blockIdx.x * 256 + threadIdx.x;
    if (i < 4 * 256) hist[i] = 0u;
    if (i == 0) { st->prefix = 0u; st->rank = NMIN; st->v = 0.0f; }
}

// ------------------------------------------------- loss + first histogram ---
// One thread = 4 consecutive pixels (one b128 per channel). 19 channel rows
// held in VGPRs (~80 regs) so logits are read exactly once, nontemporally.
__global__ __launch_bounds__(256) void k_loss(const float* __restrict__ in,
                                              const long long* __restrict__ tgt,
                                              float* __restrict__ loss,
                                              unsigned int* __restrict__ hist0) {
    __shared__ unsigned int lh[256];
    lh[threadIdx.x] = 0u;
    __syncthreads();

    int g = blockIdx.x * 256 + threadIdx.x;    // group id in [0, NG)
    int b = g >> 16;                           // image index (HW4 == 1<<16)
    int r = g & 0xFFFF;                        // float4 offset inside image

    const v4f* base = reinterpret_cast<const v4f*>(in)
                      + (size_t)b * (size_t)(NC * HW4) + (size_t)r;

    v4f x[NC];
#pragma unroll
    for (int c = 0; c < NC; ++c)
        x[c] = __builtin_nontemporal_load(base + (size_t)c * HW4);   // b128 NT

    v4f m = x[0];
#pragma unroll
    for (int c = 1; c < NC; ++c) m = v4max(m, x[c]);

    const long long* tp = tgt + 4 * (size_t)g;
    int t0 = (int)tp[0], t1 = (int)tp[1], t2 = (int)tp[2], t3 = (int)tp[3];

    v4f s = (v4f){0.f, 0.f, 0.f, 0.f};
    float xt0 = 0.f, xt1 = 0.f, xt2 = 0.f, xt3 = 0.f;
#pragma unroll
    for (int c = 0; c < NC; ++c) {
        v4f xc = x[c];
        s += v4exp(xc - m);
        if (t0 == c) xt0 = xc.x;   // v_cndmask selects, no scratch
        if (t1 == c) xt1 = xc.y;
        if (t2 == c) xt2 = xc.z;
        if (t3 == c) xt3 = xc.w;
    }

    v4f L;
    L.x = (t0 == IGN) ? 0.f : (__logf(s.x) + m.x - xt0);
    L.y = (t1 == IGN) ? 0.f : (__logf(s.y) + m.y - xt1);
    L.z = (t2 == IGN) ? 0.f : (__logf(s.z) + m.z - xt2);
    L.w = (t3 == IGN) ? 0.f : (__logf(s.w) + m.w - xt3);

    reinterpret_cast<v4f*>(loss)[g] = L;       // RT store: stays in 192MB L2

    // losses are >= 0 -> uint bit pattern is order-preserving
    atomicAdd(&lh[__float_as_uint(L.x) >> 24], 1u);
    atomicAdd(&lh[__float_as_uint(L.y) >> 24], 1u);
    atomicAdd(&lh[__float_as_uint(L.z) >> 24], 1u);
    atomicAdd(&lh[__float_as_uint(L.w) >> 24], 1u);
    __syncthreads();
    if (lh[threadIdx.x]) atomicAdd(&hist0[threadIdx.x], lh[threadIdx.x]);
}

// ------------------------------------------------ scan one radix digit ------
__global__ void k_scan(const unsigned int* __restrict__ hist,
                       SelState* __restrict__ st, int pass) {
    unsigned int r = st->rank;
    unsigned int cum = 0;
    int bsel = 0;
    for (int bin = 255; bin >= 0; --bin) {     // descending value order
        unsigned int c = hist[bin];
        if (cum + c > r) { bsel = bin; break; }
        cum += c;
    }
    st->rank = r - cum;
    unsigned int pfx = (st->prefix << 8) | (unsigned int)bsel;
    st->prefix = pfx;
    if (pass == 3) st->v = __uint_as_float(pfx);  // full 32-bit key -> exact value
}

// ------------------------------------- histogram next digit under prefix ----
__global__ __launch_bounds__(256) void k_refine(const float* __restrict__ loss,
                                                const SelState* __restrict__ st,
                                                unsigned int* __restrict__ histp,
                                                int shift) {
    __shared__ unsigned int lh[256];
    lh[threadIdx.x] = 0u;
    __syncthreads();
    unsigned int pfx = st->prefix;             // uniform scalar load
    const v4f* lp = reinterpret_cast<const v4f*>(loss);
    int stride = gridDim.x * blockDim.x;
    int tid = blockIdx.x * blockDim.x + threadIdx.x;

#ifdef HAVE_ASYNC
    // double-buffered async global->LDS staging; each lane owns its 16B slot,
    // so only the issuing wave's ASYNCcnt gates the LDS read (no barrier).
    __shared__ __align__(16) float stage[2][256 * 4];
    float* slot0 = &stage[0][threadIdx.x * 4];
    float* slot1 = &stage[1][threadIdx.x * 4];
    if (tid < NG) async_ld16(lp + tid, slot0);
    int k = 0;
    for (int i = tid; i < NG; i += stride, ++k) {
        int inext = i + stride;
        if (inext < NG) async_ld16(lp + inext, (k & 1) ? slot0 : slot1);
        wait_async(inext < NG ? 1 : 0);
        v4f L = *reinterpret_cast<const v4f*>((k & 1) ? slot1 : slot0);
#else
    for (int i = tid; i < NG; i += stride) {
        if (i + stride < NG)
            __builtin_prefetch(lp + i + stride, 0, 1);   // global_prefetch_b8
        v4f L = lp[i];
#endif
        unsigned int k0 = __float_as_uint(L.x), k1 = __float_as_uint(L.y);
        unsigned int k2 = __float_as_uint(L.z), k3 = __float_as_uint(L.w);
        if ((k0 >> shift) == pfx) atomicAdd(&lh[(k0 >> (shift - 8)) & 255u], 1u);
        if ((k1 >> shift) == pfx) atomicAdd(&lh[(k1 >> (shift - 8)) & 255u], 1u);
        if ((k2 >> shift) == pfx) atomicAdd(&lh[(k2 >> (shift - 8)) & 255u], 1u);
        if ((k3 >> shift) == pfx) atomicAdd(&lh[(k3 >> (shift - 8)) & 255u], 1u);
    }
    __syncthreads();
    if (lh[threadIdx.x]) atomicAdd(&histp[threadIdx.x], lh[threadIdx.x]);
}

// --------------------------- sums/counts for both OHEM branches -------------
__global__ __launch_bounds__(256) void k_sums(const float* __restrict__ loss,
                                              const SelState* __restrict__ st,
                                              double* __restrict__ bs1,
                                              unsigned int* __restrict__ bc1,
                                              double* __restrict__ bs2,
                                              unsigned int* __restrict__ bc2) {
    float v = st->v;
    double s1 = 0.0, s2 = 0.0;
    unsigned int c1 = 0, c2 = 0;
    const v4f* lp = reinterpret_cast<const v4f*>(loss);
    int stride = gridDim.x * blockDim.x;
    int tid = blockIdx.x * blockDim.x + threadIdx.x;

#ifdef HAVE_ASYNC
    __shared__ __align__(16) float stage[2][256 * 4];
    float* slot0 = &stage[0][threadIdx.x * 4];
    float* slot1 = &stage[1][threadIdx.x * 4];
    if (tid < NG) async_ld16(lp + tid, slot0);
    int k = 0;
    for (int i = tid; i < NG; i += stride, ++k) {
        int inext = i + stride;
        if (inext < NG) async_ld16(lp + inext, (k & 1) ? slot0 : slot1);
        wait_async(inext < NG ? 1 : 0);
        v4f L = *reinterpret_cast<const v4f*>((k & 1) ? slot1 : slot0);
#else
    for (int i = tid; i < NG; i += stride) {
        if (i + stride < NG)
            __builtin_prefetch(lp + i + stride, 0, 1);   // global_prefetch_b8
        v4f L = lp[i];
#endif
        float f[4] = {L.x, L.y, L.z, L.w};
#pragma unroll
        for (int j = 0; j < 4; ++j) {
            if (f[j] > THRESH) { s1 += (double)f[j]; c1++; }
            if (f[j] > v)      { s2 += (double)f[j]; c2++; }
        }
    }
    __shared__ double  ds1[256], ds2[256];
    __shared__ unsigned int uc1[256], uc2[256];
    int t = threadIdx.x;
    ds1[t] = s1; ds2[t] = s2; uc1[t] = c1; uc2[t] = c2;
    __syncthreads();
    for (int off = 128; off > 0; off >>= 1) {
        if (t < off) {
            ds1[t] += ds1[t + off]; ds2[t] += ds2[t + off];
            uc1[t] += uc1[t + off]; uc2[t] += uc2[t + off];
        }
        __syncthreads();
    }
    if (t == 0) {
        bs1[blockIdx.x] = ds1[0]; bc1[blockIdx.x] = uc1[0];
        bs2[blockIdx.x] = ds2[0]; bc2[blockIdx.x] = uc2[0];
    }
}

// ------------------------------------------------------------- finalize -----
__global__ __launch_bounds__(256) void k_final(const SelState* __restrict__ st,
                                               const double* __restrict__ bs1,
                                               const unsigned int* __restrict__ bc1,
                                               const double* __restrict__ bs2,
                                               const unsigned int* __restrict__ bc2,
                                               float* __restrict__ out) {
    __shared__ double  ds1[256], ds2[256];
    __shared__ unsigned int uc1[256], uc2[256];
    int t = threadIdx.x;
    double s1 = 0.0, s2 = 0.0;
    unsigned int c1 = 0, c2 = 0;
    for (int i = t; i < NBLK; i += 256) {
        s1 += bs1[i]; c1 += bc1[i];
        s2 += bs2[i]; c2 += bc2[i];
    }
    ds1[t] = s1; ds2[t] = s2; uc1[t] = c1; uc2[t] = c2;
    __syncthreads();
    for (int off = 128; off > 0; off >>= 1) {
        if (t < off) {
            ds1[t] += ds1[t + off]; ds2[t] += ds2[t + off];
            uc1[t] += uc1[t + off]; uc2[t] += uc2[t + off];
        }
        __syncthreads();
    }
    if (t == 0) {
        float v = st->v;
        double res;
        if (v > THRESH) {
            // keep everything strictly above THRESH
            res = ds1[0] / (double)uc1[0];
        } else {
            // top-NMIN: all strictly > v, plus (NMIN - cnt) ties equal to v
            res = (ds2[0] + (double)(NMIN - (long long)uc2[0]) * (double)v)
                  / (double)NMIN;
        }
        out[0] = (float)res;
    }
}

// ---------------------------------------------------------------------------
extern "C" void kernel_launch(void* const* d_in, const int* in_sizes, int n_in,
                              void* d_out, int out_size, void* d_ws, size_t ws_size,
                              hipStream_t stream) {
    const float*     in  = (const float*)d_in[0];      // (16,19,512,512) f32
    const long long* tgt = (const long long*)d_in[1];  // (16,512,512)    i64
    float*           out = (float*)d_out;              // scalar f32

    char* ws = (char*)d_ws;                            // needs ~16.03 MB
    float*        loss = (float*)(ws + LOSS_OFF);
    unsigned int* hist = (unsigned int*)(ws + HIST_OFF);   // [4][256]
    SelState*     st   = (SelState*)(ws + STATE_OFF);
    double*       bs1  = (double*)(ws + PART_OFF);
    double*       bs2  = bs1 + NBLK;
    unsigned int* bc1  = (unsigned int*)(bs2 + NBLK);
    unsigned int* bc2  = bc1 + NBLK;

    k_init  <<<4, 256, 0, stream>>>(hist, st);
    k_loss  <<<NG / 256, 256, 0, stream>>>(in, tgt, loss, hist);
    k_scan  <<<1, 1, 0, stream>>>(hist + 0 * 256, st, 0);
    k_refine<<<NBLK, 256, 0, stream>>>(loss, st, hist + 1 * 256, 24);
    k_scan  <<<1, 1, 0, stream>>>(hist + 1 * 256, st, 1);
    k_refine<<<NBLK, 256, 0, stream>>>(loss, st, hist + 2 * 256, 16);
    k_scan  <<<1, 1, 0, stream>>>(hist + 2 * 256, st, 2);
    k_refine<<<NBLK, 256, 0, stream>>>(loss, st, hist + 3 * 256, 8);
    k_scan  <<<1, 1, 0, stream>>>(hist + 3 * 256, st, 3);
    k_sums  <<<NBLK, 256, 0, stream>>>(loss, st, bs1, bc1, bs2, bc2);
    k_final <<<1, 256, 0, stream>>>(st, bs1, bc1, bs2, bc2, out);
}